// ConvolutionStack_19224273617411
// MI455X (gfx1250) — compile-verified
//
#include <hip/hip_runtime.h>

typedef __attribute__((ext_vector_type(2))) float v2f;
typedef __attribute__((ext_vector_type(4))) float v4f;
typedef __attribute__((ext_vector_type(8))) float v8f;

#define CCH    128      // residual channels
#define TLEN   8192
#define BATCH  4
#define NLAYER 20
#define NT     64       // time tile per block (4 WMMA N-tiles)
#define NTT    4        // N tiles
// LDS activation tiles are stored in WMMA-B fragment order:
//   tile[k4][col][kw]  (k4 = K/4 group, col = timestep, kw = K%4)
// so a lane's B fragment (K rows r0,r0+1 at one column) is one aligned
// ds_load_b64, and all 32 lanes x 2 words hit 64 distinct banks (no padding).
#define KGRP   (CCH / 4)          // 32 K-groups per 128 rows
#define TILEF  (CCH * NT)         // floats per tile (8192)

// ---------------------------------------------------------------------------
// Weight repack kernels: rewrite weights into per-lane WMMA A-fragment order
// so the GEMM inner loop is a single coalesced 8-byte load per fragment.
// A-matrix 16x4 f32 layout (ISA 7.12.2): lanes 0-15 -> M=lane, VGPR0=K0,
// VGPR1=K1 ; lanes 16-31 -> M=lane-16, VGPR0=K2, VGPR1=K3.
// ---------------------------------------------------------------------------

// W'[l][o][kk]: kk<128 -> w_conv[l][o][kk][0] (tap at t-d), else [kk-128][1]
// dst flat: ((((l*16 + m)*64 + ks)*32 + lane)*2 + j)
__global__ __launch_bounds__(256) void pack_wconv(const float* __restrict__ w,
                                                  float* __restrict__ dst) {
  int e = blockIdx.x * 256 + threadIdx.x;
  if (e >= NLAYER * 16 * 64 * 64) return;
  int j    = e & 1;
  int lane = (e >> 1) & 31;
  int ks   = (e >> 6) & 63;
  int m    = (e >> 12) & 15;
  int l    = e >> 16;
  int o  = m * 16 + (lane & 15);
  int kk = ks * 4 + (lane >> 4) * 2 + j;
  int c  = kk & 127;
  int k  = kk >> 7;
  dst[e] = w[((size_t)(l * 256 + o) * 128 + c) * 2 + k];
}

// dst flat: ((((l*8 + m)*32 + ks)*32 + lane)*2 + j)
__global__ __launch_bounds__(256) void pack_wout(const float* __restrict__ w,
                                                 float* __restrict__ dst) {
  int e = blockIdx.x * 256 + threadIdx.x;
  if (e >= NLAYER * 8 * 32 * 64) return;
  int j    = e & 1;
  int lane = (e >> 1) & 31;
  int ks   = (e >> 6) & 31;
  int m    = (e >> 11) & 7;
  int l    = e >> 14;
  int o = m * 16 + (lane & 15);
  int c = ks * 4 + (lane >> 4) * 2 + j;
  dst[e] = w[(size_t)(l * 128 + o) * 128 + c];
}

// ---------------------------------------------------------------------------
// One WaveNet layer, fully fused: z = W'*[x(t-d);x(t)]+b ; g = tanh*sigmoid ;
// s = Wout*g + bout ; skip = s ; x' = s + x.
// Block: 256 threads (8 waves), one batch, 64-timestep tile, all channels.
// GEMM1: wave w owns z M-tiles {w, w+8} (rows 16w.. and 128+16w..) x 4 N-tiles
// -> the tanh row and its sigmoid partner live in matching accumulator regs.
// GEMM2: wave w owns s M-tile w x 4 N-tiles.
// ---------------------------------------------------------------------------
__global__ __launch_bounds__(256)
void wavenet_layer(const float* __restrict__ xin,
                   float* __restrict__ xout,
                   float* __restrict__ skip,        // layer slice [B,C,T]
                   const float* __restrict__ packA, // layer slice [16][64][32][2]
                   const float* __restrict__ packO, // layer slice [8][32][32][2]
                   const float* __restrict__ bconv, // [256]
                   const float* __restrict__ bout,  // [128]
                   int d) {
  __shared__ float xs_halo[TILEF];  // x[t0-d .. t0-d+63], fragment order
  __shared__ float xs_main[TILEF];  // x[t0 .. t0+63],     fragment order
  __shared__ float gs[TILEF];       // gated activations,  fragment order

  const int tid  = threadIdx.x;
  const int lane = tid & 31;
  const int wv   = tid >> 5;                       // wave 0..7
  const int bid  = blockIdx.x;
  const int b    = bid / (TLEN / NT);
  const int t0   = (bid % (TLEN / NT)) * NT;
  const size_t xbase = (size_t)b * CCH * TLEN;

  // ---- stage x tiles to LDS in fragment order ----------------------------
  // item = (k4, col): gather 4 channel rows, one ds_store_b128.
  #pragma unroll
  for (int it = 0; it < (KGRP * NT) / 256; ++it) {   // 8 iterations
    int e   = it * 256 + tid;
    int k4  = e >> 6, col = e & 63;
    size_t g = xbase + (size_t)(k4 * 4) * TLEN + t0 + col;
    // main tile
    v4f vm;
    vm.x = xin[g];
    vm.y = xin[g + TLEN];
    vm.z = xin[g + 2 * TLEN];
    vm.w = xin[g + 3 * TLEN];
    *(v4f*)&xs_main[(k4 * NT + col) * 4] = vm;
    // halo tile (zero-fill left edge)
    int ts = t0 - d + col;
    v4f vh;
    if (ts >= 0) {
      size_t h = xbase + (size_t)(k4 * 4) * TLEN + ts;
      vh.x = xin[h];
      vh.y = xin[h + TLEN];
      vh.z = xin[h + 2 * TLEN];
      vh.w = xin[h + 3 * TLEN];
    } else {
      vh = (v4f){0.f, 0.f, 0.f, 0.f};
    }
    *(v4f*)&xs_halo[(k4 * NT + col) * 4] = vh;
  }
  __syncthreads();

  const int half = lane >> 4;   // fragment half
  const int nlo  = lane & 15;   // N column within tile
  const int r0   = half * 2;    // first K row this lane reads for B frags

  // ---- GEMM 1: z[256 x 64] = W'[256 x 256] * Xcat[256 x 64] --------------
  v8f acc[2][NTT] = {};         // [m-tile: wv, wv+8][n-tile 0..3]
  const float* pa0 = packA + (size_t)(wv)     * 64 * 64;
  const float* pa1 = packA + (size_t)(wv + 8) * 64 * 64;

  #pragma unroll 1
  for (int hs = 0; hs < 2; ++hs) {                 // halo rows then main rows
    const float* xsrc = hs ? xs_main : xs_halo;
    const float* wa0  = pa0 + hs * 32 * 64;
    const float* wa1  = pa1 + hs * 32 * 64;
    for (int ks = 0; ks < 32; ++ks) {
      v2f a0 = *(const v2f*)(wa0 + (ks * 32 + lane) * 2);
      v2f a1 = *(const v2f*)(wa1 + (ks * 32 + lane) * 2);
      const float* src = xsrc + ks * (NT * 4);
      v2f bf[NTT];
      #pragma unroll
      for (int ni = 0; ni < NTT; ++ni)
        bf[ni] = *(const v2f*)&src[(ni * 16 + nlo) * 4 + r0];
      #pragma unroll
      for (int ni = 0; ni < NTT; ++ni) {
        acc[0][ni] = __builtin_amdgcn_wmma_f32_16x16x4_f32(
            false, a0, false, bf[ni], (short)0, acc[0][ni], false, false);
        acc[1][ni] = __builtin_amdgcn_wmma_f32_16x16x4_f32(
            false, a1, false, bf[ni], (short)0, acc[1][ni], false, false);
      }
    }
  }

  // ---- bias + gating (register-local: tanh rows pair with sigmoid rows) --
  // C/D layout: VGPR r -> M = r (lanes 0-15) or r+8 (lanes 16-31)
  #pragma unroll
  for (int ni = 0; ni < NTT; ++ni) {
    #pragma unroll
    for (int r = 0; r < 8; ++r) {
      int row = wv * 16 + half * 8 + r;
      float zt = acc[0][ni][r] + bconv[row];          // filter half
      float zs = acc[1][ni][r] + bconv[128 + row];    // gate half
      float e2 = __expf(2.0f * zt);
      float th = (e2 - 1.0f) / (e2 + 1.0f);           // tanh
      float sg = 1.0f / (1.0f + __expf(-zs));         // sigmoid
      gs[((row >> 2) * NT + ni * 16 + nlo) * 4 + (row & 3)] = th * sg;
    }
  }
  __syncthreads();

  // ---- GEMM 2: s[128 x 64] = Wout[128 x 128] * g[128 x 64] ---------------
  v8f sa[NTT] = {};
  const float* po = packO + (size_t)wv * 32 * 64;
  for (int ks = 0; ks < 32; ++ks) {
    v2f a = *(const v2f*)(po + (ks * 32 + lane) * 2);
    const float* src = gs + ks * (NT * 4);
    v2f bf[NTT];
    #pragma unroll
    for (int ni = 0; ni < NTT; ++ni)
      bf[ni] = *(const v2f*)&src[(ni * 16 + nlo) * 4 + r0];
    #pragma unroll
    for (int ni = 0; ni < NTT; ++ni) {
      sa[ni] = __builtin_amdgcn_wmma_f32_16x16x4_f32(
          false, a, false, bf[ni], (short)0, sa[ni], false, false);
    }
  }

  // ---- epilogue: skip (nontemporal) + residual x' ------------------------
  #pragma unroll
  for (int ni = 0; ni < NTT; ++ni) {
    #pragma unroll
    for (int r = 0; r < 8; ++r) {
      int row = wv * 16 + half * 8 + r;
      int col = ni * 16 + nlo;
      float s = sa[ni][r] + bout[row];
      size_t idx = xbase + (size_t)row * TLEN + t0 + col;
      __builtin_nontemporal_store(s, &skip[idx]);            // write-once
      xout[idx] = s + xs_main[((row >> 2) * NT + col) * 4 + (row & 3)];
    }
  }
}

// ---------------------------------------------------------------------------
extern "C" void kernel_launch(void* const* d_in, const int* in_sizes, int n_in,
                              void* d_out, int out_size, void* d_ws, size_t ws_size,
                              hipStream_t stream) {
  const float* x      = (const float*)d_in[0];
  const float* w_conv = (const float*)d_in[1];
  const float* b_conv = (const float*)d_in[2];
  const float* w_out  = (const float*)d_in[3];
  const float* b_out  = (const float*)d_in[4];

  const size_t PLANE = (size_t)BATCH * CCH * TLEN;   // 4,194,304 floats
  float* out_x = (float*)d_out;                      // final residual output
  float* skips = (float*)d_out + PLANE;              // [L,B,C,T]

  float* ws0   = (float*)d_ws;                       // x ping buffer
  float* packA = ws0 + PLANE;                        // L * 256*256
  float* packO = packA + (size_t)NLAYER * 256 * 256; // L * 128*128

  pack_wconv<<<(NLAYER * 16 * 64 * 64 + 255) / 256, 256, 0, stream>>>(w_conv, packA);
  pack_wout <<<(NLAYER * 8 * 32 * 64 + 255) / 256, 256, 0, stream>>>(w_out, packO);

  static const int DIL[NLAYER] = {1, 2, 4, 8, 16, 32, 64, 128, 256, 512,
                                  1, 2, 4, 8, 16, 32, 64, 128, 256, 512};

  // Ping-pong: x_k lives in ws0 (k odd) or out_x (k even); x_20 -> out_x.
  for (int l = 0; l < NLAYER; ++l) {
    const float* rin = (l == 0) ? x : ((l & 1) ? ws0 : out_x);
    float* wbuf = ((l + 1) & 1) ? ws0 : out_x;
    wavenet_layer<<<BATCH * (TLEN / NT), 256, 0, stream>>>(
        rin, wbuf, skips + (size_t)l * PLANE,
        packA + (size_t)l * 256 * 256,
        packO + (size_t)l * 128 * 128,
        b_conv + l * 256, b_out + l * 128, DIL[l]);
  }
}